// MLDEL_5_52269751992450
// MI455X (gfx1250) — compile-verified
//
#include <hip/hip_runtime.h>
#include <hip/hip_bf16.h>

// ---------------------------------------------------------------------------
// Problem constants (from the reference)
// ---------------------------------------------------------------------------
#define NN     100000
#define EE     1600000
#define EMB    256
#define DD     128
#define BB     4096
#define LL     50
#define NCLS   10

#define RB     2            // 16-row blocks per wave (32 rows); 100000 % 32 == 0

typedef float v2f __attribute__((ext_vector_type(2)));
typedef float v8f __attribute__((ext_vector_type(8)));

// ---------------------------------------------------------------------------
// Fp32 WMMA GEMM:  out[M x 128] = A[M x K] @ W[K x 128] + bias
// One wave per 32-row block: RB=2 A fragments x 8 column tiles -> 16 v8f
// accumulators. Each B fragment (2 loads) now feeds RB WMMAs, so the loop
// issues 20 loads per 16 WMMAs instead of 16 loads per 8 (was loadcnt-bound).
//
// Lane layouts per CDNA5 ISA 7.12.2:
//   A (16x4):  lanes 0-15 hold M=0..15; VGPR0/1 = K={0,1} (lanes<16), K={2,3}
//   B (4x16):  mirrored: lanes 0-15 = N cols, halves select K pairs
//   C/D (16x16): VGPR i = row i (lanes 0-15) / row i+8 (lanes 16-31), N=lane%16
// ---------------------------------------------------------------------------
__global__ __launch_bounds__(32) void wmma_gemm_bias_f32(
    const float* __restrict__ A, const float* __restrict__ W,
    const float* __restrict__ bias, float* __restrict__ out,
    int K)
{
    const int m0    = blockIdx.x * (16 * RB);
    const int lane  = threadIdx.x;              // 0..31, wave32
    const int laneM = lane & 15;                // row (A) / col (B,D)
    const int kOff  = (lane >> 4) << 1;         // 0 or 2
    const int rowHi = (lane >> 4) << 3;         // 0 or 8 for C/D store

    v8f acc[RB][8];
#pragma unroll
    for (int r = 0; r < RB; ++r)
#pragma unroll
        for (int t = 0; t < 8; ++t) acc[r][t] = (v8f)0.0f;

    const float* Arow[RB];
#pragma unroll
    for (int r = 0; r < RB; ++r)
        Arow[r] = A + (size_t)(m0 + r * 16 + laneM) * K;

    for (int k = 0; k < K; k += 4) {
        v2f a[RB];
#pragma unroll
        for (int r = 0; r < RB; ++r) {
            a[r].x = Arow[r][k + kOff];
            a[r].y = Arow[r][k + kOff + 1];
        }
#pragma unroll
        for (int t = 0; t < 8; ++t) {
            const float* Wp = W + (size_t)(k + kOff) * DD + (t << 4) + laneM;
            v2f b;
            b.x = Wp[0];
            b.y = Wp[DD];
#pragma unroll
            for (int r = 0; r < RB; ++r)
                acc[r][t] = __builtin_amdgcn_wmma_f32_16x16x4_f32(
                    false, a[r], false, b, (short)0, acc[r][t], false, false);
        }
    }

#pragma unroll
    for (int r = 0; r < RB; ++r) {
#pragma unroll
        for (int t = 0; t < 8; ++t) {
            const int n  = (t << 4) + laneM;
            const float bv = bias[n];
            float* op = out + (size_t)(m0 + r * 16 + rowHi) * DD + n;
#pragma unroll
            for (int i = 0; i < 8; ++i)
                op[(size_t)i * DD] = acc[r][t][i] + bv;
        }
    }
}

// ---------------------------------------------------------------------------
// SpMM scatter: out[dst[e]] += w[e] * h[src[e]]   (128-float messages)
// One wave per edge; each lane owns 4 contiguous floats (float4 gather +
// 4x global_atomic_add_f32). Both h and out are L2-resident (51 MB each).
// ---------------------------------------------------------------------------
__global__ __launch_bounds__(256) void spmm_scatter(
    const int* __restrict__ src, const int* __restrict__ dst,
    const float* __restrict__ w, const float* __restrict__ h,
    float* __restrict__ out, int E)
{
    const int e    = (blockIdx.x * 256 + threadIdx.x) >> 5;
    const int lane = threadIdx.x & 31;
    if (e >= E) return;

    const int   s  = src[e];
    const int   d  = dst[e];
    const float wt = w[e];

    const float4 v = ((const float4*)(h + (size_t)s * DD))[lane];
    float* op = out + (size_t)d * DD + (lane << 2);
    unsafeAtomicAdd(op + 0, v.x * wt);
    unsafeAtomicAdd(op + 1, v.y * wt);
    unsafeAtomicAdd(op + 2, v.z * wt);
    unsafeAtomicAdd(op + 3, v.w * wt);
}

// ---------------------------------------------------------------------------
// x1 = a0 * p + x1 * (1 - p)   (p = n_param per node), in place on x1
// ---------------------------------------------------------------------------
__global__ __launch_bounds__(256) void combine_gate(
    const float* __restrict__ a0, float* __restrict__ x1,
    const float* __restrict__ np_, int total)
{
    const int i = blockIdx.x * 256 + threadIdx.x;
    if (i >= total) return;
    const float p = np_[i >> 7];
    x1[i] = a0[i] * p + x1[i] * (1.0f - p);
}

// ---------------------------------------------------------------------------
// Head: per batch element b (one block):
//   gather 50 rows of x2, append row-sum (51 rows x 128)
//   t1   = sel @ weight2 + bias2          (51 x 64)
//   sel2 = flat(t1) @ lin1_w + lin1_b     (128)   -> d_out[BB + b*128 ..]
//   pre  = sel2 @ classifier + cbias      (10)    -> argmax -> d_out[b]
// ---------------------------------------------------------------------------
__global__ __launch_bounds__(256) void head_kernel(
    const float* __restrict__ x2, const int* __restrict__ bidx,
    const float* __restrict__ w2, const float* __restrict__ b2,
    const float* __restrict__ l1w, const float* __restrict__ l1b,
    const float* __restrict__ cls, const float* __restrict__ clsb,
    float* __restrict__ out)
{
    __shared__ float sel[(LL + 1) * DD];     // 51*128
    __shared__ float t1[(LL + 1) * 64];      // 51*64 = 3264
    __shared__ float sel2[DD];
    __shared__ float pre[16];

    const int b   = blockIdx.x;
    const int tid = threadIdx.x;

    // gather 50 rows
    for (int idx = tid; idx < LL * DD; idx += 256) {
        const int l = idx >> 7, d = idx & (DD - 1);
        const int node = bidx[b * LL + l];
        sel[idx] = x2[(size_t)node * DD + d];
    }
    __syncthreads();

    // row 50 = sum over rows
    for (int d = tid; d < DD; d += 256) {
        float s = 0.0f;
        for (int l = 0; l < LL; ++l) s += sel[l * DD + d];
        sel[LL * DD + d] = s;
    }
    __syncthreads();

    // t1[l][e] = sel[l] . weight2[:,e] + bias2[e]
    for (int idx = tid; idx < (LL + 1) * 64; idx += 256) {
        const int l = idx >> 6, e = idx & 63;
        float s = b2[e];
        const float* sr = &sel[l * DD];
        for (int d = 0; d < DD; ++d) s += sr[d] * w2[d * 64 + e];
        t1[idx] = s;
    }
    __syncthreads();

    // sel2[j] = flat(t1) . lin1_w[:,j] + lin1_b[j]
    if (tid < DD) {
        float s = l1b[tid];
        for (int i = 0; i < (LL + 1) * 64; ++i)
            s += t1[i] * l1w[(size_t)i * DD + tid];
        sel2[tid] = s;
        out[(size_t)BB + (size_t)b * DD + tid] = s;
    }
    __syncthreads();

    // classifier + argmax
    if (tid < NCLS) {
        float s = clsb[tid];
        for (int j = 0; j < DD; ++j) s += sel2[j] * cls[j * NCLS + tid];
        pre[tid] = s;
    }
    __syncthreads();
    if (tid == 0) {
        int best = 0; float bv = pre[0];
        for (int c = 1; c < NCLS; ++c)
            if (pre[c] > bv) { bv = pre[c]; best = c; }
        out[b] = (float)best;
    }
}

// ---------------------------------------------------------------------------
// Launch: ws holds 3 ping-pong [N x 128] fp32 buffers (3 * 51.2 MB).
// ---------------------------------------------------------------------------
extern "C" void kernel_launch(void* const* d_in, const int* in_sizes, int n_in,
                              void* d_out, int out_size, void* d_ws, size_t ws_size,
                              hipStream_t stream)
{
    const float* A        = (const float*)d_in[0];   // [N,256]
    const int*   esrc     = (const int*)  d_in[1];
    const int*   edst     = (const int*)  d_in[2];
    const float* ew       = (const float*)d_in[3];
    const int*   bidx     = (const int*)  d_in[4];   // [B,1,L]
    const float* w_gc1    = (const float*)d_in[5];
    const float* b_gc1    = (const float*)d_in[6];
    const float* w_gc2    = (const float*)d_in[7];
    const float* b_gc2    = (const float*)d_in[8];
    const float* nprm     = (const float*)d_in[9];
    const float* Lin1     = (const float*)d_in[10];
    const float* Lin1b    = (const float*)d_in[11];
    const float* weight2  = (const float*)d_in[12];
    const float* bias2    = (const float*)d_in[13];
    const float* lin1w    = (const float*)d_in[14];
    const float* lin1b    = (const float*)d_in[15];
    const float* cls      = (const float*)d_in[16];
    const float* clsb     = (const float*)d_in[17];
    float* out = (float*)d_out;

    const size_t ND = (size_t)NN * DD;               // 12.8M floats
    float* buf0 = (float*)d_ws;                      // h1 / x2
    float* buf1 = buf0 + ND;                         // a0 / h2
    float* buf2 = buf1 + ND;                         // x1

    const int gemmBlocks = NN / (16 * RB);           // 3125 (N % 32 == 0)
    const int spmmBlocks = (EE + 7) / 8;             // 8 waves (edges) / block
    const int ewBlocks   = (int)((ND + 255) / 256);

    // a0 = A @ Lin1 + Lin1_bias
    wmma_gemm_bias_f32<<<gemmBlocks, 32, 0, stream>>>(A, Lin1, Lin1b, buf1, EMB);
    // h1 = A @ w_gc1 + b_gc1
    wmma_gemm_bias_f32<<<gemmBlocks, 32, 0, stream>>>(A, w_gc1, b_gc1, buf0, EMB);

    // x1 = spmm(h1)
    hipMemsetAsync(buf2, 0, ND * sizeof(float), stream);
    spmm_scatter<<<spmmBlocks, 256, 0, stream>>>(esrc, edst, ew, buf0, buf2, EE);

    // x1 = a0*p + x1*(1-p)
    combine_gate<<<ewBlocks, 256, 0, stream>>>(buf1, buf2, nprm, (int)ND);

    // h2 = x1 @ w_gc2 + b_gc2
    wmma_gemm_bias_f32<<<gemmBlocks, 32, 0, stream>>>(buf2, w_gc2, b_gc2, buf1, DD);

    // x2 = spmm(h2)
    hipMemsetAsync(buf0, 0, ND * sizeof(float), stream);
    spmm_scatter<<<spmmBlocks, 256, 0, stream>>>(esrc, edst, ew, buf1, buf0, EE);

    // head: gather + small GEMMs + classifier/argmax
    head_kernel<<<BB, 256, 0, stream>>>(buf0, bidx, weight2, bias2,
                                        lin1w, lin1b, cls, clsb, out);
}